// NeRFRenderer_36017595744318
// MI455X (gfx1250) — compile-verified
//
#include <hip/hip_runtime.h>
#include <math.h>

// ---------------------------------------------------------------------------
// Problem sizes (compile-time constants from the reference)
// ---------------------------------------------------------------------------
#define NQ   65536          // N_RAYS * N_STEPS
#define NV   6890           // vertices
#define NVP  6896           // padded to multiple of 16
#define NVF  (NV * 3)       // raw vertex dwords
#define NT   (NVP / 16)     // 431 vertex tiles
#define NF   13776
#define NVT  7576

typedef __attribute__((ext_vector_type(2))) float v2f;
typedef __attribute__((ext_vector_type(8))) float v8f;

#if defined(__HIP_DEVICE_COMPILE__) && \
    __has_builtin(__builtin_amdgcn_global_load_async_to_lds_b128) && \
    __has_builtin(__builtin_amdgcn_global_load_async_to_lds_b32)
#define USE_ASYNC_LDS 1
typedef int v4i __attribute__((vector_size(16)));
typedef __attribute__((address_space(1))) v4i* gv4i_p;   // prints as "__device__"
typedef __attribute__((address_space(3))) v4i* lv4i_p;   // prints as "__shared__"
typedef __attribute__((address_space(1))) int* gi32_p;
typedef __attribute__((address_space(3))) int* li32_p;
#else
#define USE_ASYNC_LDS 0
#endif

__device__ __forceinline__ float clamp01(float x) {
    return fminf(fmaxf(x, 0.f), 1.f);
}

// ---------------------------------------------------------------------------
// Fused pipeline, one block = 256 threads = 8 waves = 256 queries.
//
// Phase 1 (WMMA 1-NN):  d2(m,n) = |v_n|^2 - 2*dot(q_m, v_n), argmin over n.
//   Each wave owns TWO 16-query A fragments (32 queries), sweeps all 431
//   16-vertex B tiles from LDS with a software-pipelined fetch, issuing two
//   V_WMMA_F32_16X16X4_F32 per tile. Per-lane running (min,argmin) for 8
//   M-rows; final argmin over N via 4-step shfl_xor in each 16-lane half.
//
// Phase 2 (per-thread): candidate-face pick, closest point on triangle,
//   signed distance, barycentric UV, sigmoid, clip. Triangle vertices are
//   read from the LDS-resident copy.
// ---------------------------------------------------------------------------
__global__ __launch_bounds__(256)
void nerf_fused_kernel(const float* __restrict__ pts,
                       const float* __restrict__ verts,
                       const float* __restrict__ fn,
                       const float* __restrict__ fc,
                       const float* __restrict__ uv,
                       const int*   __restrict__ vf,
                       const int*   __restrict__ mesh_f,
                       const int*   __restrict__ uv_f,
                       float*       __restrict__ out) {
    __shared__ __align__(16) float sverts[NVP * 3];  // raw xyz stream, 82,752 B
    __shared__ float ssq[NVP];                       // |v|^2, 27,584 B
    __shared__ int   sidx[256];                      // per-query nearest vertex

    const int tid    = threadIdx.x;
    const int qblock = blockIdx.x * 256;

    // ---- stage vertices into LDS (gfx1250 async global->LDS, ASYNCcnt) ---
#if USE_ASYNC_LDS
    {
        const int nchunk = NVF / 4;                 // 5167 b128 chunks
        for (int c = tid; c < nchunk; c += 256) {
            __builtin_amdgcn_global_load_async_to_lds_b128(
                (gv4i_p)(verts + 4 * c), (lv4i_p)(sverts + 4 * c), 0, 0);
        }
        if (tid < NVF - nchunk * 4) {               // 2 tail dwords
            __builtin_amdgcn_global_load_async_to_lds_b32(
                (gi32_p)(verts + nchunk * 4 + tid),
                (li32_p)(sverts + nchunk * 4 + tid), 0, 0);
        }
    }
#else
    for (int i = tid; i < NVF; i += 256) sverts[i] = verts[i];
#endif
    // zero the pad region (disjoint from async targets)
    for (int i = NVF + tid; i < NVP * 3; i += 256) sverts[i] = 0.f;

    // warm the cache with this block's query points while staging is in flight
    __builtin_prefetch(pts + 3 * (qblock + tid), 0, 0);

#if USE_ASYNC_LDS
    asm volatile("s_wait_asynccnt 0x0" ::: "memory");
#endif
    __syncthreads();

    // |v|^2 (pad verts get +huge so they never win the argmin)
    for (int i = tid; i < NVP; i += 256) {
        const float x = sverts[3 * i], y = sverts[3 * i + 1], z = sverts[3 * i + 2];
        ssq[i] = (i < NV) ? (x * x + y * y + z * z) : 3.0e38f;
    }
    __syncthreads();

    // ---- phase 1: WMMA nearest-vertex sweep ------------------------------
    const int  lane  = tid & 31;
    const int  wave  = tid >> 5;
    const int  m     = lane & 15;
    const bool hi    = lane >= 16;
    const int  qbase = qblock + wave * 32;          // 2 tiles of 16 per wave

    // A fragments: lanes 0-15 -> K=0/1, lanes 16-31 -> K=2/3 (K=3 zero pad)
    v2f A[2];
#pragma unroll
    for (int q = 0; q < 2; ++q) {
        const float* qp = pts + 3 * (qbase + q * 16 + m);
        const float qx = qp[0], qy = qp[1], qz = qp[2];
        A[q].x = hi ? qz : qx;
        A[q].y = hi ? 0.f : qy;
    }

    float bd[2][8];
    int   bi[2][8];
#pragma unroll
    for (int q = 0; q < 2; ++q)
#pragma unroll
        for (int k = 0; k < 8; ++k) { bd[q][k] = 3.4e38f; bi[q][k] = 0; }

    // software-pipelined B-tile fetch
    float nx = sverts[3 * m], ny = sverts[3 * m + 1], nz = sverts[3 * m + 2];
    float nsq = ssq[m];

    for (int t = 0; t < NT; ++t) {
        const float cx = nx, cy = ny, cz = nz, csq = nsq;
        const int   cn = t * 16 + m;

        const int tn = (t + 1 < NT) ? t + 1 : t;    // prefetch next tile
        const int nn = tn * 16 + m;
        nx = sverts[3 * nn]; ny = sverts[3 * nn + 1]; nz = sverts[3 * nn + 2];
        nsq = ssq[nn];

        v2f B;
        B.x = hi ? cz : cx;                         // K = 0 | 2
        B.y = hi ? 0.f : cy;                        // K = 1 | 3 (zero pad)

#pragma unroll
        for (int q = 0; q < 2; ++q) {
            v8f C = {};
            C = __builtin_amdgcn_wmma_f32_16x16x4_f32(
                    false, A[q], false, B, (short)0, C, false, false);
#pragma unroll
            for (int k = 0; k < 8; ++k) {
                const float d2    = __builtin_fmaf(-2.f, C[k], csq);
                const bool better = d2 < bd[q][k];
                bd[q][k] = better ? d2 : bd[q][k];
                bi[q][k] = better ? cn : bi[q][k];
            }
        }
    }

    // argmin over the 16 vertex columns in each half-wave (branchless)
#pragma unroll
    for (int q = 0; q < 2; ++q)
#pragma unroll
        for (int k = 0; k < 8; ++k) {
            float d  = bd[q][k];
            int   ix = bi[q][k];
#pragma unroll
            for (int mask = 1; mask <= 8; mask <<= 1) {
                const float od   = __shfl_xor(d, mask, 32);
                const int   oi   = __shfl_xor(ix, mask, 32);
                const bool  take = (od < d) || ((od == d) && (oi < ix));
                d  = take ? od : d;
                ix = take ? oi : ix;
            }
            if (lane == 0 || lane == 16)
                sidx[wave * 32 + q * 16 + (hi ? 8 : 0) + k] = ix;
        }
    __syncthreads();

    // ---- phase 2: closest point on triangle + signed distance + UV -------
    const int ig = qblock + tid;                    // global query id
    const float px = pts[3 * ig], py = pts[3 * ig + 1], pz = pts[3 * ig + 2];

    const int vid = sidx[tid];
    int   f  = vf[vid * 6];
    float bdist = 3.4e38f;
#pragma unroll
    for (int j = 0; j < 6; ++j) {
        const int   cf = vf[vid * 6 + j];
        const float dx = px - fc[3 * cf + 0];
        const float dy = py - fc[3 * cf + 1];
        const float dz = pz - fc[3 * cf + 2];
        const float d2 = dx * dx + dy * dy + dz * dz;
        if (d2 < bdist) { bdist = d2; f = cf; }     // strict < => first-min
    }

    const int i0 = mesh_f[3 * f + 0];
    const int i1 = mesh_f[3 * f + 1];
    const int i2 = mesh_f[3 * f + 2];
    const float ax = sverts[3 * i0], ay = sverts[3 * i0 + 1], az = sverts[3 * i0 + 2];
    const float bx = sverts[3 * i1], by = sverts[3 * i1 + 1], bz = sverts[3 * i1 + 2];
    const float cx = sverts[3 * i2], cy = sverts[3 * i2 + 1], cz = sverts[3 * i2 + 2];

    const float abx = bx - ax, aby = by - ay, abz = bz - az;
    const float acx = cx - ax, acy = cy - ay, acz = cz - az;
    const float apx = px - ax, apy = py - ay, apz = pz - az;
    const float d1 = abx * apx + aby * apy + abz * apz;
    const float d2 = acx * apx + acy * apy + acz * apz;
    const float bpx = px - bx, bpy = py - by, bpz = pz - bz;
    const float d3 = abx * bpx + aby * bpy + abz * bpz;
    const float d4 = acx * bpx + acy * bpy + acz * bpz;
    const float cpx = px - cx, cpy = py - cy, cpz = pz - cz;
    const float d5 = abx * cpx + aby * cpy + abz * cpz;
    const float d6 = acx * cpx + acy * cpy + acz * cpz;

    const float vcr = d1 * d4 - d3 * d2;
    const float vbr = d5 * d2 - d1 * d6;
    const float var = d3 * d6 - d5 * d4;

    const bool c1 = (d1 <= 0.f) && (d2 <= 0.f);
    const bool c2 = (d3 >= 0.f) && (d4 <= d3);
    const bool c3 = (d6 >= 0.f) && (d5 < d6);
    const bool c4 = (vcr <= 0.f) && (d1 >= 0.f) && (d3 <= 0.f);
    const bool c5 = (vbr <= 0.f) && (d2 >= 0.f) && (d6 <= 0.f);
    const bool c6 = (var <= 0.f) && (d4 >= d3) && (d6 <= d5);
    const bool interior = !(c1 || c2 || c3 || c4 || c5 || c6);

    const float v_ab = d1 / (c4 ? (d1 - d3) : 1.f);
    const float w_ac = d2 / (c5 ? (d2 - d6) : 1.f);
    const float w_bc = (d4 - d3) / (c6 ? (d4 - d3 + (d5 - d6)) : 1.f);
    const float inv  = 1.f / (interior ? (var + vbr + vcr) : 1.f);
    const float v_in = vbr * inv;
    const float w_in = vcr * inv;

    float prx = ax + v_in * abx + w_in * acx;
    float pry = ay + v_in * aby + w_in * acy;
    float prz = az + v_in * abz + w_in * acz;
    float b0 = 1.f - v_in - w_in, b1 = v_in, b2 = w_in;

    // sequential overwrite in the reference's where-chain order
    if (c1) { prx = ax; pry = ay; prz = az; b0 = 1.f; b1 = 0.f; b2 = 0.f; }
    if (c2) { prx = bx; pry = by; prz = bz; b0 = 0.f; b1 = 1.f; b2 = 0.f; }
    if (c3) { prx = cx; pry = cy; prz = cz; b0 = 0.f; b1 = 0.f; b2 = 1.f; }
    if (c4) { prx = ax + v_ab * abx; pry = ay + v_ab * aby; prz = az + v_ab * abz;
              b0 = 1.f - v_ab; b1 = v_ab; b2 = 0.f; }
    if (c5) { prx = ax + w_ac * acx; pry = ay + w_ac * acy; prz = az + w_ac * acz;
              b0 = 1.f - w_ac; b1 = 0.f; b2 = w_ac; }
    if (c6) { prx = bx + w_bc * (cx - bx); pry = by + w_bc * (cy - by);
              prz = bz + w_bc * (cz - bz);
              b0 = 0.f; b1 = 1.f - w_bc; b2 = w_bc; }

    const float ddx = prx - px, ddy = pry - py, ddz = prz - pz;
    const float dist = sqrtf(ddx * ddx + ddy * ddy + ddz * ddz);

    const float dn = fn[3 * f + 0] * (px - prx) +
                     fn[3 * f + 1] * (py - pry) +
                     fn[3 * f + 2] * (pz - prz);
    const float sgn = (dn > 0.f) ? 1.f : ((dn < 0.f) ? -1.f : 0.f);
    const float sd  = sgn * dist;

    const int u0 = uv_f[3 * f + 0];
    const int u1 = uv_f[3 * f + 1];
    const int u2 = uv_f[3 * f + 2];
    const float uvx = b0 * uv[2 * u0 + 0] + b1 * uv[2 * u1 + 0] + b2 * uv[2 * u2 + 0];
    const float uvy = b0 * uv[2 * u0 + 1] + b1 * uv[2 * u1 + 1] + b2 * uv[2 * u2 + 1];

    const float dsig = 1.f / (1.f + __expf(-10.f * sd));

    out[3 * ig + 0] = clamp01(uvx);
    out[3 * ig + 1] = clamp01(uvy);
    out[3 * ig + 2] = clamp01(dsig);
    out[3 * NQ + ig] = sd;
}

// ---------------------------------------------------------------------------
// Launch
// ---------------------------------------------------------------------------
extern "C" void kernel_launch(void* const* d_in, const int* in_sizes, int n_in,
                              void* d_out, int out_size, void* d_ws, size_t ws_size,
                              hipStream_t stream) {
    const float* pts    = (const float*)d_in[0];
    const float* verts  = (const float*)d_in[1];
    const float* fnorm  = (const float*)d_in[2];
    const float* fcent  = (const float*)d_in[3];
    const float* uv     = (const float*)d_in[4];
    const int*   vf     = (const int*)d_in[5];
    const int*   mesh_f = (const int*)d_in[6];
    const int*   uv_f   = (const int*)d_in[7];
    float*       out    = (float*)d_out;
    (void)d_ws; (void)ws_size; (void)in_sizes; (void)n_in; (void)out_size;

    // 256 blocks x 256 threads; each block owns 256 queries end-to-end.
    nerf_fused_kernel<<<NQ / 256, 256, 0, stream>>>(
        pts, verts, fnorm, fcent, uv, vf, mesh_f, uv_f, out);
}